// GNN_72645076844628
// MI455X (gfx1250) — compile-verified
//
#include <hip/hip_runtime.h>

// ---------------------------------------------------------------------------
// 2-layer GCN on MI455X (gfx1250, wave32).
//   out = D^-1/2 (A+I) D^-1/2 (relu(D^-1/2 (A+I) D^-1/2 (x W1) + b1) W2) + b2
// GEMMs: V_WMMA_F32_16X16X4_F32 (true fp32 matrix path; GEMM is ~5 GFLOP,
// nowhere near the bottleneck, so no precision sacrifice).
// Tile staging: GLOBAL_LOAD_ASYNC_TO_LDS_B128 (CDNA5 async path, ASYNCcnt).
// Edge scatter (~2.5 GB irregular traffic) is the HBM-roofline limiter.
// ---------------------------------------------------------------------------

typedef float v2f __attribute__((ext_vector_type(2)));
typedef float v4f __attribute__((ext_vector_type(4)));
typedef float v8f __attribute__((ext_vector_type(8)));

#define NODES 100000
#define EDGES 1600000
#define FIN   128
#define FHID  128
#define FOUT  64

// ---- degree / normalization ------------------------------------------------

__global__ void init_deg_kernel(float* __restrict__ deg, int n) {
    int i = blockIdx.x * blockDim.x + threadIdx.x;
    if (i < n) deg[i] = 1.0f;               // implicit self-loop
}

__global__ void count_deg_kernel(const int* __restrict__ dst, float* deg, int e) {
    int i = blockIdx.x * blockDim.x + threadIdx.x;
    if (i < e) atomicAdd(&deg[dst[i]], 1.0f);  // exact: counts < 2^24
}

__global__ void rsqrt_kernel(float* __restrict__ deg, int n) {
    int i = blockIdx.x * blockDim.x + threadIdx.x;
    if (i < n) deg[i] = rsqrtf(deg[i]);     // deg >= 1 always (self-loop)
}

// ---- fp32 WMMA GEMM: C[m0:m0+16, :cols] = (relu?)A[m0:m0+16, 0:128] @ W ----
// One block = one 16-row tile. One wave = one 16-column tile.
// blockDim.x = 32 * (cols/16).

__global__ __launch_bounds__(256) void gemm_wmma_kernel(
    const float* __restrict__ A, const float* __restrict__ W,
    float* __restrict__ C, int cols, int applyRelu) {
    __shared__ float tx[16][FIN + 4];       // +4 pad: 16-lane LDS reads hit 16 banks

    const int tid  = threadIdx.x;
    const int lane = tid & 31;
    const int wave = tid >> 5;
    const int m0   = blockIdx.x * 16;

    if (!applyRelu) {
        // The 16x128 tile is one contiguous 8KB span of A (ldA == FIN).
        // CDNA5 async copy: memory -> LDS directly, no VGPR round-trip.
        // Each lane moves 16B per issue; tracked by ASYNCcnt.
        for (int e = tid * 4; e < 16 * FIN; e += blockDim.x * 4) {
            int r = e >> 7, c = e & (FIN - 1);
            unsigned ldsOff = (unsigned)(unsigned long long)&tx[r][c];
            const float* gp = A + ((size_t)m0 * FIN + e);
            asm volatile("global_load_async_to_lds_b128 %0, %1, off"
                         :: "v"(ldsOff), "v"(gp) : "memory");
        }
        asm volatile("s_wait_asynccnt 0x0" ::: "memory");
    } else {
        // Layer 2: ReLU fused into staging (saves a 51 MB elementwise pass),
        // so data must pass through VALU.
        for (int e = tid; e < 16 * FIN; e += blockDim.x) {
            int r = e >> 7, c = e & (FIN - 1);
            tx[r][c] = fmaxf(A[(size_t)m0 * FIN + e], 0.0f);
        }
    }
    __syncthreads();

    // ISA 7.12.2 fragment layouts (wave32):
    //   A 16x4 f32: lanes 0-15 -> M=lane, K={k0,k0+1}; lanes 16-31 -> K={k0+2,k0+3}
    //   B 4x16 f32: mirrored K split, N = lane&15
    //   C 16x16 f32: VGPR r -> M=r (lanes 0-15) / M=r+8 (lanes 16-31), N=lane&15
    const int row  = lane & 15;
    const int koff = (lane >> 4) << 1;      // 0 or 2
    const int col  = wave * 16 + (lane & 15);

    const float* Wp = W + (koff * cols + col);   // 32-bit offsets from here on
    v8f acc = {};
    int woff = 0;                                // = k0 * cols
    for (int k0 = 0; k0 < FIN; k0 += 4, woff += 4 * cols) {
        v2f a, b;
        a.x = tx[row][k0 + koff];
        a.y = tx[row][k0 + koff + 1];
        b.x = Wp[woff];
        b.y = Wp[woff + cols];
        // (neg_a, A, neg_b, B, c_mod, C, reuse_a, reuse_b)
        acc = __builtin_amdgcn_wmma_f32_16x16x4_f32(
            false, a, false, b, (short)0, acc, false, false);
    }

    const int rbase = (lane >> 4) << 3;     // 0 or 8
    float* Cp = C + ((size_t)(m0 + rbase) * cols + col);
    for (int r = 0; r < 8; ++r)
        Cp[r * cols] = acc[r];
}

// ---- agg init: bias + self-loop term (writes EVERY output element) ---------
// Vectorized float4: F is a multiple of 4, rows stay aligned.

__global__ void bias_selfloop_kernel(const float* __restrict__ t,
                                     const float* __restrict__ dinv,
                                     const float* __restrict__ bias,
                                     float* __restrict__ agg,
                                     int total4, int logF) {
    int q = blockIdx.x * blockDim.x + threadIdx.x;   // index in float4 units
    if (q >= total4) return;
    int idx = q << 2;
    int i = idx >> logF;
    int c = idx & ((1 << logF) - 1);
    float di = dinv[i];
    float n2 = di * di;                              // self-loop norm
    v4f tv = *(const v4f*)(t + idx);
    const v4f bv = *(const v4f*)(bias + c);
    v4f o;
    o.x = bv.x + tv.x * n2;
    o.y = bv.y + tv.y * n2;
    o.z = bv.z + tv.z * n2;
    o.w = bv.w + tv.w * n2;
    *(v4f*)(agg + idx) = o;
}

// ---- edge scatter: agg[dst] += h[src] * dinv[src]*dinv[dst] ----------------
// One wave per edge; lane loads F/32 consecutive floats as one vector
// (coalesced 512B/256B row gather), then f32 global atomics. This is the
// memory-roofline-dominant kernel.

template <int F>
__global__ void edge_scatter_kernel(const float* __restrict__ h,
                                    const float* __restrict__ dinv,
                                    const int* __restrict__ src,
                                    const int* __restrict__ dst,
                                    float* __restrict__ agg,
                                    int e) {
    constexpr int VW = F / 32;               // 4 for F=128, 2 for F=64
    typedef float vwf __attribute__((ext_vector_type(VW)));

    int wid  = (blockIdx.x * blockDim.x + threadIdx.x) >> 5;
    int lane = threadIdx.x & 31;
    if (wid >= e) return;
    int s = src[wid], d = dst[wid];
    float nrm = dinv[s] * dinv[d];

    const float* hs = h   + (size_t)s * F + lane * VW;
    float*       ad = agg + (size_t)d * F + lane * VW;
    vwf v = *(const vwf*)hs;                 // one b128/b64 load per lane
#pragma unroll
    for (int j = 0; j < VW; ++j)
        atomicAdd(&ad[j], v[j] * nrm);       // global_atomic_add_f32
}

// ---------------------------------------------------------------------------

extern "C" void kernel_launch(void* const* d_in, const int* in_sizes, int n_in,
                              void* d_out, int out_size, void* d_ws, size_t ws_size,
                              hipStream_t stream) {
    const float* x   = (const float*)d_in[0];            // [N,128]
    const int*   ei  = (const int*)  d_in[1];            // [2,E]
    const float* W1  = (const float*)d_in[2];            // [128,128]
    const float* b1  = (const float*)d_in[3];            // [128]
    const float* W2  = (const float*)d_in[4];            // [128,64]
    const float* b2  = (const float*)d_in[5];            // [64]
    float*       out = (float*)d_out;                    // [N,64]

    const int* srcI = ei;                                // edge_index[0]
    const int* dstI = ei + EDGES;                        // edge_index[1]

    // Workspace layout (~103 MB): dinv | bufA (N*128) | bufB (N*128)
    float* dinv = (float*)d_ws;
    float* bufA = dinv + ((NODES + 255) & ~255);
    float* bufB = bufA + (size_t)NODES * FHID;

    // 1) normalization coefficients
    init_deg_kernel<<<(NODES + 255) / 256, 256, 0, stream>>>(dinv, NODES);
    count_deg_kernel<<<(EDGES + 255) / 256, 256, 0, stream>>>(dstI, dinv, EDGES);
    rsqrt_kernel<<<(NODES + 255) / 256, 256, 0, stream>>>(dinv, NODES);

    // 2) layer 1: h = x @ W1   (fp32 WMMA; async-to-LDS staging)
    gemm_wmma_kernel<<<NODES / 16, 256, 0, stream>>>(x, W1, bufA, FHID, 0);
    // agg1 = b1 + h*dinv^2 (self-loop), then scatter E edges
    bias_selfloop_kernel<<<(NODES * FHID / 4 + 255) / 256, 256, 0, stream>>>(
        bufA, dinv, b1, bufB, NODES * FHID / 4, 7);
    edge_scatter_kernel<FHID><<<EDGES / 8, 256, 0, stream>>>(
        bufA, dinv, srcI, dstI, bufB, EDGES);

    // 3) layer 2: t = relu(agg1) @ W2  (ReLU fused into LDS staging)
    gemm_wmma_kernel<<<NODES / 16, 128, 0, stream>>>(bufB, W2, bufA, FOUT, 1);
    bias_selfloop_kernel<<<(NODES * FOUT / 4 + 255) / 256, 256, 0, stream>>>(
        bufA, dinv, b2, out, NODES * FOUT / 4, 6);
    edge_scatter_kernel<FOUT><<<EDGES / 8, 256, 0, stream>>>(
        bufA, dinv, srcI, dstI, out, EDGES);
}